// PointnetFPModule_59081570124918
// MI455X (gfx1250) — compile-verified
//
#include <hip/hip_runtime.h>

typedef __attribute__((ext_vector_type(2))) float v2f;
typedef __attribute__((ext_vector_type(8))) float v8f;

#define B_    16
#define NQ    4096
#define MK    1024
#define C1_   256
#define C2_   256
#define CIN   512
#define COUT  256
#define NCOLS (B_ * NQ) /* 65536 */

// X is stored K-pair-interleaved: element (k, j) lives at
//   Xp[(k>>1) * 2*NCOLS + 2*j + (k&1)]
// so the WMMA B-fragment (rows k,k+1 at column j) is one aligned 8-byte load.

// ---------------------------------------------------------------------------
// Pass 1: brute-force 3-NN per query point. Known points staged in LDS (12KB).
// ---------------------------------------------------------------------------
__global__ void k_three_nn(const float* __restrict__ unknown,
                           const float* __restrict__ known,
                           int* __restrict__ idx_out,
                           float* __restrict__ w_out) {
  __shared__ float pts[MK * 3];
  const int b = blockIdx.x >> 4;                       // 16 chunks per batch
  const int n = ((blockIdx.x & 15) << 8) + threadIdx.x;
  const float* kb = known + (size_t)b * MK * 3;
  for (int i = threadIdx.x; i < MK * 3; i += 256) pts[i] = kb[i];
  __syncthreads();

  const float* q = unknown + ((size_t)b * NQ + n) * 3;
  const float qx = q[0], qy = q[1], qz = q[2];
  float d0 = 1e30f, d1 = 1e30f, d2 = 1e30f;
  int i0 = 0, i1 = 0, i2 = 0;
  for (int i = 0; i < MK; ++i) {
    const float dx = qx - pts[i * 3 + 0];
    const float dy = qy - pts[i * 3 + 1];
    const float dz = qz - pts[i * 3 + 2];
    const float d = dx * dx + dy * dy + dz * dz;
    if (d < d0)      { d2 = d1; i2 = i1; d1 = d0; i1 = i0; d0 = d; i0 = i; }
    else if (d < d1) { d2 = d1; i2 = i1; d1 = d;  i1 = i; }
    else if (d < d2) { d2 = d;  i2 = i; }
  }
  const float r0 = 1.0f / (d0 + 1e-8f);
  const float r1 = 1.0f / (d1 + 1e-8f);
  const float r2 = 1.0f / (d2 + 1e-8f);
  const float rs = 1.0f / (r0 + r1 + r2);
  const size_t o = ((size_t)b * NQ + n) * 3;
  idx_out[o + 0] = i0; idx_out[o + 1] = i1; idx_out[o + 2] = i2;
  w_out[o + 0] = r0 * rs; w_out[o + 1] = r1 * rs; w_out[o + 2] = r2 * rs;
}

// ---------------------------------------------------------------------------
// Pass 2: build Xp (K-pair-interleaved, K=512). Channel pairs 0..127 =
// interpolated feats, pairs 128..255 = unknow_feats. Gathers served by L2
// (known_feats = 16MB, L2 = 192MB); stores are coalesced b64.
// ---------------------------------------------------------------------------
__global__ void k_build_x(const float* __restrict__ known_feats,
                          const float* __restrict__ unknow_feats,
                          const int* __restrict__ idx_in,
                          const float* __restrict__ w_in,
                          float* __restrict__ Xp) {
  const int b  = blockIdx.x >> 6;
  const int n0 = (blockIdx.x & 63) << 6;   // 64 columns per block
  const int cl = threadIdx.x & 63;         // column within chunk
  const int pr = threadIdx.x >> 6;         // 0..3 pair phase
  const int n  = n0 + cl;
  const size_t j = (size_t)b * NQ + n;     // global GEMM column
  const size_t o = j * 3;
  const int   i0 = idx_in[o + 0], i1 = idx_in[o + 1], i2 = idx_in[o + 2];
  const float w0 = w_in[o + 0],  w1 = w_in[o + 1],  w2 = w_in[o + 2];

  const float* kf = known_feats + (size_t)b * C2_ * MK;
  for (int p = pr; p < C2_ / 2; p += 4) {          // channel pair p -> c=2p,2p+1
    const float* r0 = kf + (size_t)(2 * p) * MK;
    const float* r1 = r0 + MK;
    v2f v;
    v.x = w0 * r0[i0] + w1 * r0[i1] + w2 * r0[i2];
    v.y = w0 * r1[i0] + w1 * r1[i1] + w2 * r1[i2];
    *(v2f*)(Xp + (size_t)p * 2 * NCOLS + 2 * j) = v;
  }
  const float* uf = unknow_feats + (size_t)b * C1_ * NQ;
  for (int p = pr; p < C1_ / 2; p += 4) {          // pairs 128..255
    const float* r0 = uf + (size_t)(2 * p) * NQ;
    v2f v;
    v.x = r0[n];
    v.y = r0[NQ + n];
    *(v2f*)(Xp + (size_t)(C2_ / 2 + p) * 2 * NCOLS + 2 * j) = v;
  }
}

// ---------------------------------------------------------------------------
// Pass 3/5: H = W(256xK) * X(K x NCOLS) with V_WMMA_F32_16X16X4_F32.
// Block = 8 waves; wave w owns M-tiles {32w, 32w+16} x N-tiles {j0, j0+16}.
// A 16x4 frag: lane m=l&15 holds K={2g,2g+1} (g=l>>4): one b64 load from W.
// B 4x16 frag: one b64 load from pair-interleaved Xp (128B/half-wave).
// Inner loop: 4 b64 loads -> 4 WMMAs.
// ---------------------------------------------------------------------------
__global__ void k_gemm_wmma(const float* __restrict__ W,
                            const float* __restrict__ Xp,
                            float* __restrict__ H,
                            int K) {
  const int lane = threadIdx.x & 31;
  const int wv   = threadIdx.x >> 5;   // 0..7 (wave32)
  const int g    = lane >> 4;          // 0/1
  const int ml   = lane & 15;
  const int m0   = wv * 32;
  const int j0   = blockIdx.x * 32;

  v8f acc00 = {}, acc01 = {}, acc10 = {}, acc11 = {};
  const float* Wr0 = W + (size_t)(m0 + ml) * K;
  const float* Wr1 = W + (size_t)(m0 + 16 + ml) * K;
  const float* Xb0 = Xp + 2 * (size_t)(j0 + ml);        // +ka*NCOLS per step
  const float* Xb1 = Xb0 + 32;                          // j0+16 tile

#pragma unroll 4
  for (int k = 0; k < K; k += 4) {
    const int ka = k + 2 * g;                           // even
    v2f a0 = *(const v2f*)(Wr0 + ka);
    v2f a1 = *(const v2f*)(Wr1 + ka);
    v2f b0 = *(const v2f*)(Xb0 + (size_t)ka * NCOLS);
    v2f b1 = *(const v2f*)(Xb1 + (size_t)ka * NCOLS);
    acc00 = __builtin_amdgcn_wmma_f32_16x16x4_f32(false, a0, false, b0, (short)0, acc00, false, false);
    acc01 = __builtin_amdgcn_wmma_f32_16x16x4_f32(false, a0, false, b1, (short)0, acc01, false, false);
    acc10 = __builtin_amdgcn_wmma_f32_16x16x4_f32(false, a1, false, b0, (short)0, acc10, false, false);
    acc11 = __builtin_amdgcn_wmma_f32_16x16x4_f32(false, a1, false, b1, (short)0, acc11, false, false);
  }

  // D layout: VGPR v holds row m = v + 8*g, col = lane&15 (ISA 7.12.2)
  float* Hb = H + (size_t)j0 + (size_t)ml;
#pragma unroll
  for (int v = 0; v < 8; ++v) {
    const int mr0 = m0 + v + 8 * g;
    const int mr1 = m0 + 16 + v + 8 * g;
    Hb[(size_t)mr0 * NCOLS + 0]  = acc00[v];
    Hb[(size_t)mr0 * NCOLS + 16] = acc01[v];
    Hb[(size_t)mr1 * NCOLS + 0]  = acc10[v];
    Hb[(size_t)mr1 * NCOLS + 16] = acc11[v];
  }
}

// ---------------------------------------------------------------------------
// Pass 4/6: per-channel mean/var over 65536 cols (one WG per channel,
// fixed-order tree reduction -> deterministic). Emits fused scale/bias.
// ---------------------------------------------------------------------------
__global__ void k_bn_stats(const float* __restrict__ H,
                           const float* __restrict__ gamma,
                           const float* __restrict__ beta,
                           float* __restrict__ scale,
                           float* __restrict__ bias) {
  __shared__ float ssum[256];
  __shared__ float ssq[256];
  const int c = blockIdx.x;
  const float* row = H + (size_t)c * NCOLS;
  float s = 0.f, q = 0.f;
  for (int j = threadIdx.x; j < NCOLS; j += 256) {
    const float v = row[j];
    s += v; q += v * v;
  }
  ssum[threadIdx.x] = s;
  ssq[threadIdx.x]  = q;
  __syncthreads();
  for (int off = 128; off > 0; off >>= 1) {
    if ((int)threadIdx.x < off) {
      ssum[threadIdx.x] += ssum[threadIdx.x + off];
      ssq[threadIdx.x]  += ssq[threadIdx.x + off];
    }
    __syncthreads();
  }
  if (threadIdx.x == 0) {
    const float inv = 1.0f / (float)NCOLS;
    const float mean = ssum[0] * inv;
    const float var  = ssq[0] * inv - mean * mean;  // biased, matches jnp.var
    const float sc   = gamma[c] * rsqrtf(var + 1e-5f);
    scale[c] = sc;
    bias[c]  = beta[c] - mean * sc;
  }
}

// BN + ReLU on H (row-major C x NCOLS) -> Xp2 (pair-interleaved, for GEMM2).
// One thread per (channel-pair, column): 2 coalesced b32 reads, 1 b64 store.
__global__ void k_bn_relu_pair(const float* __restrict__ H,
                               const float* __restrict__ scale,
                               const float* __restrict__ bias,
                               float* __restrict__ Xp2) {
  const size_t t = (size_t)blockIdx.x * 256 + threadIdx.x;
  const int j = (int)(t & (NCOLS - 1));
  const int p = (int)(t >> 16);          // channel pair 0..127
  const int c = 2 * p;
  const float v0 = H[(size_t)c * NCOLS + j] * scale[c] + bias[c];
  const float v1 = H[(size_t)(c + 1) * NCOLS + j] * scale[c + 1] + bias[c + 1];
  v2f v;
  v.x = v0 > 0.f ? v0 : 0.f;
  v.y = v1 > 0.f ? v1 : 0.f;
  *(v2f*)(Xp2 + (size_t)p * 2 * NCOLS + 2 * (size_t)j) = v;
}

// Final BN + ReLU, writing (B, 256, n) layout
__global__ void k_bn_relu_out(const float* __restrict__ H,
                              const float* __restrict__ scale,
                              const float* __restrict__ bias,
                              float* __restrict__ out) {
  const size_t t = (size_t)blockIdx.x * 256 + threadIdx.x;
  const int n = (int)(t & (NQ - 1));
  const int o = (int)((t >> 12) & 255);
  const int b = (int)(t >> 20);
  const float v = H[(size_t)o * NCOLS + (size_t)b * NQ + n] * scale[o] + bias[o];
  out[t] = v > 0.f ? v : 0.f;
}

// ---------------------------------------------------------------------------
extern "C" void kernel_launch(void* const* d_in, const int* in_sizes, int n_in,
                              void* d_out, int out_size, void* d_ws, size_t ws_size,
                              hipStream_t stream) {
  (void)in_sizes; (void)n_in; (void)out_size; (void)ws_size;
  const float* unknown      = (const float*)d_in[0];
  const float* known        = (const float*)d_in[1];
  const float* unknow_feats = (const float*)d_in[2];
  const float* known_feats  = (const float*)d_in[3];
  const float* W1 = (const float*)d_in[4];
  const float* g1 = (const float*)d_in[5];
  const float* b1 = (const float*)d_in[6];
  const float* W2 = (const float*)d_in[7];
  const float* g2 = (const float*)d_in[8];
  const float* b2 = (const float*)d_in[9];

  // Workspace layout. Xp (128MB) is consumed by GEMM1, then its region is
  // reused for the pair-interleaved GEMM2 input Xp2 (64MB).
  float* Xp  = (float*)d_ws;                        // 512*65536 f32 = 128MB
  float* H1  = Xp + (size_t)CIN  * NCOLS;           // 256*65536 f32 = 64MB
  float* H2  = H1 + (size_t)COUT * NCOLS;           // 64MB
  int*   idx = (int*)(H2 + (size_t)COUT * NCOLS);   // B*n*3 ints
  float* wgt = (float*)(idx + (size_t)B_ * NQ * 3); // B*n*3 f32
  float* st  = wgt + (size_t)B_ * NQ * 3;           // scale1,bias1,scale2,bias2
  float* Xp2 = Xp;                                  // reuse (GEMM1 is done)

  k_three_nn<<<B_ * (NQ / 256), 256, 0, stream>>>(unknown, known, idx, wgt);
  k_build_x <<<B_ * (NQ / 64),  256, 0, stream>>>(known_feats, unknow_feats, idx, wgt, Xp);

  k_gemm_wmma<<<NCOLS / 32, 256, 0, stream>>>(W1, Xp, H1, CIN);
  k_bn_stats <<<COUT, 256, 0, stream>>>(H1, g1, b1, st, st + 256);
  k_bn_relu_pair<<<(COUT / 2) * (NCOLS / 256), 256, 0, stream>>>(H1, st, st + 256, Xp2);

  k_gemm_wmma<<<NCOLS / 32, 256, 0, stream>>>(W2, Xp2, H2, COUT);
  k_bn_stats <<<COUT, 256, 0, stream>>>(H2, g2, b2, st + 512, st + 768);
  k_bn_relu_out<<<(COUT * NCOLS) / 256, 256, 0, stream>>>(H2, st + 512, st + 768,
                                                          (float*)d_out);
}